// Spectrum_48378511622592
// MI455X (gfx1250) — compile-verified
//
#include <hip/hip_runtime.h>

typedef __attribute__((ext_vector_type(2))) float v2f;
typedef __attribute__((ext_vector_type(8))) float v8f;

#define BAND    80          // input bins covered per 16-wide k tile (actual need ~66)
#define NCHUNK  (BAND / 4)  // 20 WMMA K=4 steps
#define WSTRIDE 82          // LDS row stride in floats: even (8B-aligned b64 reads)
                            // and 18l mod 64 distinct for l=0..15 (conflict-free)

// Bin edge m of grid wl[0..n-1], m in [0, n]. e_first/e_last are the
// half-bin-extrapolated outer edges, precomputed by the caller. Branchless.
__device__ __forceinline__ float edge_at(const float* __restrict__ wl, int n,
                                         float e_first, float e_last, int m) {
    int mc = m;
    mc = mc < 1 ? 1 : mc;
    mc = mc > n - 1 ? n - 1 : mc;
    float v = 0.5f * (wl[mc - 1] + wl[mc]);
    v = (m <= 0) ? e_first : v;
    v = (m >= n) ? e_last : v;
    return v;
}

// One block per 16-wide k tile; 4 waves sweep all n tiles.
// Requires M >= BAND (harness: M = 16384).
__global__ __launch_bounds__(128) void rebin_wmma_f32(
    const float* __restrict__ wl_in, const float* __restrict__ flux,
    const float* __restrict__ wl_out, float* __restrict__ out,
    int M, int K, int N)
{
    __shared__ __align__(16) float Wlds[16 * WSTRIDE];

    const int tid  = threadIdx.x;
    const int lane = tid & 31;
    const int wave = tid >> 5;
    const int l    = lane & 15;   // output column within tile / flux row within tile
    const int half = lane >> 4;

    const int k0 = blockIdx.x * 16;

    // Outer-edge extrapolation constants for both grids.
    const float ei_first = wl_in[0]      - 0.5f * (wl_in[1] - wl_in[0]);
    const float ei_last  = wl_in[M - 1]  + 0.5f * (wl_in[M - 1] - wl_in[M - 2]);
    const float eo_first = wl_out[0]     - 0.5f * (wl_out[1] - wl_out[0]);
    const float eo_last  = wl_out[K - 1] + 0.5f * (wl_out[K - 1] - wl_out[K - 2]);

    // Block-uniform binary search: largest m in [0, M-1] with edge_in(m) <= lower
    // output edge of this tile. All threads compute the same value (no divergence).
    const float eo_lo = edge_at(wl_out, K, eo_first, eo_last, k0);
    int lo = 0, hi = M - 1;
    while (lo < hi) {
        int mid = (lo + hi + 1) >> 1;
        float e = edge_at(wl_in, M, ei_first, ei_last, mid);
        if (e <= eo_lo) lo = mid; else hi = mid - 1;
    }
    int m_start = lo;
    if (m_start > M - BAND) m_start = M - BAND;  // whole band stays in-bounds
    if (m_start < 0) m_start = 0;
    m_start &= ~1;                               // 8B-aligned b64 flux loads

    // Phase 1: fill the 16 x BAND overlap-weight band in LDS (once per k tile).
    for (int idx = tid; idx < 16 * BAND; idx += 128) {
        const int lw = idx / BAND;   // output column within tile
        const int j  = idx % BAND;   // band position
        const int m  = m_start + j;
        const int kk = k0 + lw;
        float eo0 = edge_at(wl_out, K, eo_first, eo_last, kk);
        float eo1 = edge_at(wl_out, K, eo_first, eo_last, kk + 1);
        float ei0 = edge_at(wl_in, M, ei_first, ei_last, m     > M ? M : m);
        float ei1 = edge_at(wl_in, M, ei_first, ei_last, m + 1 > M ? M : m + 1);
        float w = fmaxf(0.0f, fminf(eo1, ei1) - fmaxf(eo0, ei0));
        w = (m < M) ? w : 0.0f;
        Wlds[lw * WSTRIDE + j] = w;
    }
    __syncthreads();

    // Per-lane output-bin normalization (column k = k0 + l for all 8 acc rows).
    const int k = k0 + l;
    const float eok  = edge_at(wl_out, K, eo_first, eo_last, k);
    const float eok1 = edge_at(wl_out, K, eo_first, eo_last, k + 1);
    const float inv_dout = 1.0f / (eok1 - eok);

    const float2* __restrict__ wrow =
        reinterpret_cast<const float2*>(&Wlds[l * WSTRIDE]);  // 8B-aligned

    // Phase 2: each wave handles n tiles wave, wave+4, ... (uniform loop bounds).
    const int ntiles = N >> 4;
    for (int nt = wave; nt < ntiles; nt += 4) {
        const int n0 = nt * 16;
        const float2* __restrict__ frow = reinterpret_cast<const float2*>(
            flux + (size_t)(n0 + l) * (size_t)M + m_start);

        v8f acc = {};
        #pragma unroll
        for (int c = 0; c < NCHUNK; ++c) {
            // A(16x4)/B(4x16) f32 layouts: lanes 0-15 own K-slices {0,1},
            // lanes 16-31 own {2,3}; jb is even -> b64 loads stay aligned.
            const int jb = 4 * c + 2 * half;
            const float2 fa = frow[jb >> 1];        // flux[n0+l][m_start+jb .. +1]
            const float2 wb = wrow[jb >> 1];        // W[m_start+jb .. +1][k0+l]
            v2f a; a.x = fa.x; a.y = fa.y;
            v2f b; b.x = wb.x; b.y = wb.y;
            acc = __builtin_amdgcn_wmma_f32_16x16x4_f32(
                false, a, false, b, (short)0, acc, false, false);
        }

        // C/D layout: VGPR g holds rows g (lanes 0-15) and g+8 (lanes 16-31).
        float* __restrict__ orow = out + (size_t)(n0 + half * 8) * (size_t)K + k;
        #pragma unroll
        for (int g = 0; g < 8; ++g) {
            orow[(size_t)g * K] = acc[g] * inv_dout;
        }
    }
}

extern "C" void kernel_launch(void* const* d_in, const int* in_sizes, int n_in,
                              void* d_out, int out_size, void* d_ws, size_t ws_size,
                              hipStream_t stream) {
    const float* wl_in  = (const float*)d_in[0];
    const float* flux   = (const float*)d_in[1];
    const float* wl_out = (const float*)d_in[2];
    float* out = (float*)d_out;

    const int M = in_sizes[0];      // 16384
    const int K = in_sizes[2];      // 4096
    const int N = in_sizes[1] / M;  // 256

    dim3 grid(K / 16);              // one block per k tile
    rebin_wmma_f32<<<grid, 128, 0, stream>>>(wl_in, flux, wl_out, out, M, K, N);
}